// OnnxIOExpert_62380105007631
// MI455X (gfx1250) — compile-verified
//
#include <hip/hip_runtime.h>
#include <stdint.h>

// Row layout constants (innermost dim = 512 f32, contiguous)
#define ROW 512
#define WAVES_PER_BLOCK 8

__device__ __forceinline__ uint32_t lds_off32(const void* p) {
    // generic pointer to LDS: low 32 bits == LDS byte offset
    return (uint32_t)(uintptr_t)p;
}

__global__ __launch_bounds__(256) void io_vm_step_kernel(
    const float* __restrict__ x, float* __restrict__ out, int nrows)
{
    __shared__ float lds[WAVES_PER_BLOCK * ROW];

    const int lane = threadIdx.x & 31;
    const int wave = threadIdx.x >> 5;
    const int row  = blockIdx.x * WAVES_PER_BLOCK + wave;
    if (row >= nrows) return;   // wave-uniform

    const float* rin  = x   + (size_t)row * ROW;
    float*       rout = out + (size_t)row * ROW;
    float*       slot = lds + wave * ROW;
    const uint32_t lbase = lds_off32(slot);

    // ---- 1) Async-DMA the whole row (2KB) global -> LDS: 4 x b128 per lane ----
    // Streaming data, zero reuse -> non-temporal at the near caches.
    #pragma unroll
    for (int j = 0; j < 4; ++j) {
        const int e = (j * 32 + lane) * 4;                   // element index, 16B aligned
        unsigned long long g = (unsigned long long)(uintptr_t)(rin + e);
        uint32_t l = lbase + (uint32_t)e * 4u;
        asm volatile("global_load_async_to_lds_b128 %0, %1, off th:TH_LOAD_NT"
                     :: "v"(l), "v"(g) : "memory");
    }
    asm volatile("s_wait_asynccnt 0" ::: "memory");

    // ---- 2) Broadcast-read per-row scalars from LDS ----
    float a[8], ioc[8];
    #pragma unroll
    for (int k = 0; k < 8; ++k) { a[k] = slot[k]; ioc[k] = slot[368 + k]; }

    const float is_open    = slot[104 + 30];   // 134
    const float is_read    = slot[104 + 31];   // 135
    const float is_prtf    = slot[104 + 33];   // 137
    const float is_exit    = slot[104 + 38];   // 142
    const float is_getchar = slot[104 + 64];   // 168
    const float is_putchar = slot[104 + 65];   // 169
    const float io_in_rdy  = slot[377];
    const float buf_len    = slot[448];
    const float buf_pos    = slot[449];

    // nibs = one-hot(|buf_pos - p| < 0.5) . buf(16x4)  ==  bounds-checked row select
    float nib0 = 0.f, nib1 = 0.f, nib2 = 0.f, nib3 = 0.f;
    {
        const float pf = rintf(buf_pos);               // round-half-even; ties -> no match
        if (fabsf(buf_pos - pf) < 0.5f && pf >= 0.0f && pf <= 15.0f) {
            const int p = (int)pf;
            nib0 = slot[384 + 4 * p + 0];
            nib1 = slot[384 + 4 * p + 1];
            nib2 = slot[384 + 4 * p + 2];
            nib3 = slot[384 + 4 * p + 3];
        }
    }

    // ---- 3) Reference math (mirrored op order) ----
    const float is_output = is_putchar + is_prtf;
    const float hs = io_in_rdy;
    const float hb = (buf_pos < buf_len) ? 1.0f : 0.0f;
    const float hi = fminf(fmaxf(hs + hb, 0.0f), 1.0f);
    const float fd = a[0];
    const float is_stdin = (fabsf(fd) < 0.5f) ? 1.0f : 0.0f;

    float soft[8];
    #pragma unroll
    for (int k = 0; k < 8; ++k) {
        const float br = (k == 0) ? nib0 : (k == 1) ? nib1 : (k == 2) ? nib2
                        : (k == 3) ? nib3 : 0.0f;
        soft[k] = hs * ioc[k] + (1.0f - hs) * hb * br;
    }

    const float need_in = is_getchar * (1.0f - hi) + is_read * is_stdin * (1.0f - hi);
    const float new_pos = buf_pos + is_getchar * (1.0f - hs) * hb
                                  + is_read * is_stdin * (1.0f - hs) * hb;
    const float new_rdy = io_in_rdy * (1.0f - is_getchar) * (1.0f - is_read * is_stdin);

    float res[8], ico[8];
    #pragma unroll
    for (int k = 0; k < 8; ++k) {
        const float rr = is_stdin * soft[k] + (1.0f - is_stdin) * 15.0f;
        res[k] = is_output * a[k] + is_getchar * soft[k] + is_exit * a[k]
               + is_open * 15.0f + is_read * rr;
        ico[k] = ioc[k] * (1.0f - is_output) + a[k] * is_output;
    }

    // ---- 4) Patch the row in LDS (lanes 0..20) ----
    if (lane < 8)        slot[16 + lane]        = res[lane];
    else if (lane < 16)  slot[368 + (lane - 8)] = ico[lane - 8];
    else if (lane == 16) slot[376]              = is_output;   // output_ready
    else if (lane == 17) slot[377]              = new_rdy;     // input_ready
    else if (lane == 18) slot[378]              = need_in;     // need_input
    else if (lane == 19) slot[379]              = is_exit;     // program_end
    else if (lane == 20) slot[449]              = new_pos;     // buf_pos

    // ensure DS patches landed before the async store engine reads LDS
    asm volatile("s_wait_dscnt 0" ::: "memory");

    // ---- 5) Async-DMA the patched row LDS -> global out: 4 x b128 per lane ----
    #pragma unroll
    for (int j = 0; j < 4; ++j) {
        const int e = (j * 32 + lane) * 4;
        unsigned long long g = (unsigned long long)(uintptr_t)(rout + e);
        uint32_t l = lbase + (uint32_t)e * 4u;
        asm volatile("global_store_async_from_lds_b128 %0, %1, off th:TH_STORE_NT"
                     :: "v"(g), "v"(l) : "memory");
    }
    // S_ENDPGM performs an implicit wait-idle, draining ASYNCcnt.
}

extern "C" void kernel_launch(void* const* d_in, const int* in_sizes, int n_in,
                              void* d_out, int out_size, void* d_ws, size_t ws_size,
                              hipStream_t stream)
{
    const float* x = (const float*)d_in[0];
    float* out = (float*)d_out;
    const int nrows  = in_sizes[0] / ROW;                 // 262144 for (4,65536,512)
    const int blocks = (nrows + WAVES_PER_BLOCK - 1) / WAVES_PER_BLOCK;
    hipLaunchKernelGGL(io_vm_step_kernel, dim3(blocks), dim3(256), 0, stream,
                       x, out, nrows);
}